// StandGraph1_50371376447881
// MI455X (gfx1250) — compile-verified
//
#include <hip/hip_runtime.h>
#include <hip/hip_bf16.h>

// GraphConv on MI455X (gfx1250, wave32):
//  K1: zero agg workspace
//  K2: edge-parallel scatter-add (global_atomic_add_f32, L2-resident agg)
//  K3: fused GEMM out = agg@W_rel^T + x@W_root^T + b, ReLU via V_WMMA_F32_16X16X4_F32

#define NFEAT 64
#define NCLASS 64
#define SW_STRIDE2 80   // float2 row stride for LDS weight tiles (160 words -> halves of the
                        // wave hit disjoint 32-bank sets: conflict-free ds_load_b64)

typedef __attribute__((ext_vector_type(2))) float v2f;
typedef __attribute__((ext_vector_type(8))) float v8f;

// ---------------- K1: zero the aggregation buffer ----------------
__global__ __launch_bounds__(256) void zero_agg_kernel(float4* __restrict__ p, int n4) {
    int i = blockIdx.x * blockDim.x + threadIdx.x;
    if (i < n4) p[i] = make_float4(0.f, 0.f, 0.f, 0.f);
}

// ---------------- K2: scatter-add neighbor features ----------------
// 16 threads per edge, each thread moves one float4 chunk (64 feats = 16 chunks).
// Lanes 0..15 of a wave read 256B contiguous from x[src] -> coalesced gather.
__global__ __launch_bounds__(256) void scatter_edges_kernel(
    const float* __restrict__ x,
    const int*   __restrict__ src,
    const int*   __restrict__ dst,
    float*       __restrict__ agg,
    int nEdges)
{
    int t = blockIdx.x * blockDim.x + threadIdx.x;
    int e = t >> 4;
    int c = t & 15;
    if (e >= nEdges) return;
    int s = src[e];
    int d = dst[e];
    const float4 v = *(const float4*)(x + (size_t)s * NFEAT + c * 4);
    float* p = agg + (size_t)d * NFEAT + c * 4;
    // no-return fp32 atomics -> global_atomic_add_f32 (STOREcnt path), resolved in L2
    unsafeAtomicAdd(p + 0, v.x);
    unsafeAtomicAdd(p + 1, v.y);
    unsafeAtomicAdd(p + 2, v.z);
    unsafeAtomicAdd(p + 3, v.w);
}

// ---------------- K3: WMMA GEMM + bias + ReLU ----------------
// Block = 256 threads = 8 waves; each wave computes a 16x64 output slab.
// A tile 16x4 f32: lane holds float2 at (m = lane&15, k = kbase..kbase+1), kbase = 2*(lane>>4)
// B tile 4x16 f32: lane holds float2 at (n = lane&15, k = kbase..kbase+1)
// C/D 16x16 f32:   vgpr r, lane -> (m = r + 8*(lane>>4), n = lane&15)
// B fragments are staged in LDS pre-paired as float2 so each lane needs a single
// ds_load_b64 per (kstep, coltile) — no cross-lane reassembly movs.
__global__ __launch_bounds__(256) void graphconv_gemm_kernel(
    const float* __restrict__ x,
    const float* __restrict__ agg,
    const float* __restrict__ W_rel,   // [NCLASS][NFEAT] row-major: W[n][k]
    const float* __restrict__ b_rel,   // [NCLASS]
    const float* __restrict__ W_root,  // [NCLASS][NFEAT]
    float*       __restrict__ out,     // [N][NCLASS]
    int nNodes)
{
    __shared__ v2f sWrel2[(NFEAT / 2) * SW_STRIDE2];   // [k2][n] -> {W[n][2k2], W[n][2k2+1]}
    __shared__ v2f sWroot2[(NFEAT / 2) * SW_STRIDE2];
    __shared__ float sBias[NCLASS];

    const int tid = threadIdx.x;

    // Stage weights into LDS as paired float2, coalesced 8B global reads.
    for (int e2 = tid; e2 < (NFEAT / 2) * NCLASS; e2 += 256) {
        int n  = e2 >> 5;    // class row of W  (NFEAT/2 == 32)
        int k2 = e2 & 31;    // feature pair index
        sWrel2[k2 * SW_STRIDE2 + n]  = *(const v2f*)(W_rel  + n * NFEAT + 2 * k2);
        sWroot2[k2 * SW_STRIDE2 + n] = *(const v2f*)(W_root + n * NFEAT + 2 * k2);
    }
    if (tid < NCLASS) sBias[tid] = b_rel[tid];
    __syncthreads();

    const int wv    = tid >> 5;       // wave in block: 0..7
    const int lane  = tid & 31;
    const int hi    = lane >> 4;      // half-wave select
    const int l16   = lane & 15;
    const int kbase = hi * 2;

    const int rowBase = (blockIdx.x * 8 + wv) * 16;
    const int mIdx    = rowBase + l16;
    // Clamp instead of predicating loads: wrong data only lands in output rows
    // we never store, and the hot loop stays free of EXEC manipulation.
    const int mClamp  = (mIdx < nNodes) ? mIdx : (nNodes - 1);

    v8f acc[4];
    acc[0] = (v8f){}; acc[1] = (v8f){}; acc[2] = (v8f){}; acc[3] = (v8f){};

    const float* aRow = agg + (size_t)mClamp * NFEAT;
    const float* xRow = x   + (size_t)mClamp * NFEAT;

    // ---- agg @ W_rel^T : 16 K-steps of 4 ----
#pragma unroll
    for (int k = 0; k < NFEAT; k += 4) {
        const v2f a = *(const v2f*)(aRow + k + kbase);            // 8B global load
        const int krow = ((k + kbase) >> 1) * SW_STRIDE2;
#pragma unroll
        for (int ct = 0; ct < 4; ++ct) {
            const v2f b = sWrel2[krow + ct * 16 + l16];           // one ds_load_b64
            acc[ct] = __builtin_amdgcn_wmma_f32_16x16x4_f32(
                false, a, false, b, (short)0, acc[ct], false, false);
        }
    }

    // ---- x @ W_root^T : 16 K-steps of 4 ----
#pragma unroll
    for (int k = 0; k < NFEAT; k += 4) {
        const v2f a = *(const v2f*)(xRow + k + kbase);
        const int krow = ((k + kbase) >> 1) * SW_STRIDE2;
#pragma unroll
        for (int ct = 0; ct < 4; ++ct) {
            const v2f b = sWroot2[krow + ct * 16 + l16];
            acc[ct] = __builtin_amdgcn_wmma_f32_16x16x4_f32(
                false, a, false, b, (short)0, acc[ct], false, false);
        }
    }

    // ---- epilogue: bias + ReLU + store ----
#pragma unroll
    for (int ct = 0; ct < 4; ++ct) {
        const int n = ct * 16 + l16;
        const float bias = sBias[n];
#pragma unroll
        for (int r = 0; r < 8; ++r) {
            const int m = rowBase + hi * 8 + r;
            if (m < nNodes) {
                float v = acc[ct][r] + bias;
                out[(size_t)m * NCLASS + n] = v > 0.f ? v : 0.f;
            }
        }
    }
}

extern "C" void kernel_launch(void* const* d_in, const int* in_sizes, int n_in,
                              void* d_out, int out_size, void* d_ws, size_t ws_size,
                              hipStream_t stream) {
    const float* x      = (const float*)d_in[0];           // [N, 64]
    const int*   adj    = (const int*)d_in[1];             // [2, E]
    const float* W_rel  = (const float*)d_in[2];           // [64, 64]
    const float* b_rel  = (const float*)d_in[3];           // [64]
    const float* W_root = (const float*)d_in[4];           // [64, 64]
    float*       out    = (float*)d_out;                   // [N, 64]
    float*       agg    = (float*)d_ws;                    // [N, 64] scratch

    const int nNodes = in_sizes[0] / NFEAT;
    const int nEdges = in_sizes[1] / 2;
    const int* src = adj;
    const int* dst = adj + nEdges;

    // K1: zero agg (float4 stores)
    {
        const int n4 = nNodes * (NFEAT / 4);
        const int blocks = (n4 + 255) / 256;
        zero_agg_kernel<<<blocks, 256, 0, stream>>>((float4*)agg, n4);
    }

    // K2: scatter-add edges (16 threads/edge)
    {
        const long long threads = (long long)nEdges * 16;
        const int blocks = (int)((threads + 255) / 256);
        scatter_edges_kernel<<<blocks, 256, 0, stream>>>(x, src, dst, agg, nEdges);
    }

    // K3: fused WMMA GEMM + bias + ReLU (128 rows per block)
    {
        const int blocks = (nNodes + 127) / 128;
        graphconv_gemm_kernel<<<blocks, 256, 0, stream>>>(x, agg, W_rel, b_rel, W_root, out, nNodes);
    }
}